// CTCLossWrapper_15393162789157
// MI455X (gfx1250) — compile-verified
//
#include <hip/hip_runtime.h>

// Problem constants from the reference
#define T_   1024
#define N_   64
#define C_   512
#define S_   128
#define L_   257        // 2*S + 1
#define NEGV (-1e9f)
#define DEPTH 4         // async LDS ring depth (power of two)

#define AS1 __attribute__((address_space(1)))
#define AS3 __attribute__((address_space(3)))

typedef __attribute__((ext_vector_type(2))) float v2f;
typedef __attribute__((ext_vector_type(8))) float v8f;
typedef int v4i __attribute__((vector_size(16)));   // matches builtin param type

__device__ __forceinline__ float laddexp(float a, float b) {
    // jnp.logaddexp for finite inputs: max + log1p(exp(-|a-b|))
    float m = fmaxf(a, b);
    float d = fabsf(a - b);
    return m + log1pf(__expf(-d));
}

// ---------------------------------------------------------------------------
// Kernel 1: logsumexp over C for every (t, n) row.
// One wave per 16-row tile. Rows are staged into LDS (padded stride 516
// floats) by GLOBAL_LOAD_ASYNC_TO_LDS_B128 with a 4-row pipeline tracked by
// ASYNCcnt; row maxes are reduced from LDS with shuffles; the sum(exp)
// reduction runs on the matrix core: V_WMMA_F32_16X16X4_F32 against a ones
// B-matrix accumulates exact f32 row sums (replicated across columns).
// A-layout (32-bit A 16x4): lanes 0-15 -> M=lane, K={k0,k0+1};
//                           lanes 16-31 -> M=lane-16, K={k0+2,k0+3}.
// C-layout: VGPR r, lanes 0-15 -> M=r ; lanes 16-31 -> M=r+8.
// ---------------------------------------------------------------------------
__global__ __launch_bounds__(32)
void ctc_lse_kernel(const float* __restrict__ preds, float* __restrict__ lse) {
    const int lane    = threadIdx.x;
    const int rowBase = blockIdx.x * 16;       // 16 rows of (T*N, C)
    const int m       = lane & 15;             // row this lane feeds into WMMA
    const int half    = lane >> 4;

    __shared__ float tile[16 * 516];           // padded row stride: 516 floats
    __shared__ float sums[16];

    // Async-copy one 2KB row into its padded LDS slot: 4 x b128 per lane.
    auto issue_row = [&](int r) {
        AS1 const char* gp =
            (AS1 const char*)(preds + (size_t)(rowBase + r) * C_) + lane * 16;
        AS3 char* lp = (AS3 char*)&tile[r * 516] + lane * 16;
        #pragma unroll
        for (int c = 0; c < 4; ++c)
            __builtin_amdgcn_global_load_async_to_lds_b128(
                (AS1 v4i*)(gp + c * 512), (AS3 v4i*)(lp + c * 512), 0, 0);
    };

    float myMax = NEGV;                        // row max for row m
    auto row_max = [&](int r) {
        const float* rowp = &tile[r * 516];
        float rmax = NEGV;
        #pragma unroll
        for (int j = 0; j < 4; ++j) {
            float4 v = *reinterpret_cast<const float4*>(rowp + 4 * lane + 128 * j);
            rmax = fmaxf(rmax, fmaxf(fmaxf(v.x, v.y), fmaxf(v.z, v.w)));
        }
        #pragma unroll
        for (int o = 16; o >= 1; o >>= 1)
            rmax = fmaxf(rmax, __shfl_xor(rmax, o, 32));
        if (m == r) myMax = rmax;              // keep only my row's max
    };

    // ---- Pass 1: async-pipelined tile staging + per-row max ----
    #pragma unroll
    for (int r = 0; r < DEPTH; ++r) issue_row(r);

    for (int r = 0; r < 16 - DEPTH; ++r) {
        __builtin_amdgcn_s_wait_asynccnt(4 * (DEPTH - 1)); // oldest row landed
        row_max(r);
        asm volatile("s_wait_dscnt 0" ::: "memory");       // reads drained
        issue_row(r + DEPTH);
    }
    __builtin_amdgcn_s_wait_asynccnt(0);
    for (int r = 16 - DEPTH; r < 16; ++r) row_max(r);
    __syncthreads();

    // ---- Pass 2: WMMA ones-reduction of exp(x - rowmax) ----
    v8f acc = {};
    v2f ones; ones[0] = 1.0f; ones[1] = 1.0f;  // B = all-ones (layout-agnostic)
    const float* myRow = &tile[m * 516];

    #pragma unroll 4
    for (int k0 = 0; k0 < C_; k0 += 4) {
        int c0 = k0 + 2 * half;
        float x0 = myRow[c0];
        float x1 = myRow[c0 + 1];
        v2f a;
        a[0] = __expf(x0 - myMax);
        a[1] = __expf(x1 - myMax);
        // (neg_a, A, neg_b, B, c_mod, C, reuse_a, reuse_b)
        acc = __builtin_amdgcn_wmma_f32_16x16x4_f32(
            false, a, false, ones, (short)0, acc, false, false);
    }

    // Extract row sums: lane 0 has rows 0..7 in acc[0..7], lane 16 rows 8..15.
    if (lane == 0) {
        #pragma unroll
        for (int r = 0; r < 8; ++r) sums[r] = acc[r];
    }
    if (lane == 16) {
        #pragma unroll
        for (int r = 0; r < 8; ++r) sums[8 + r] = acc[r];
    }
    __syncthreads();

    if (lane < 16) {
        // myMax on lane < 16 corresponds to row m == lane
        lse[rowBase + lane] = myMax + logf(sums[lane]);
    }
}

// ---------------------------------------------------------------------------
// Kernel 2: the CTC alpha recursion. One wave per batch element.
// alpha[257] lives in 9 registers per lane: s = i*32 + lane; s-1/s-2
// neighbors via shuffles (no barriers). Emission rows are prefetched 4 steps
// ahead into an LDS ring by GLOBAL_LOAD_ASYNC_TO_LDS_B128; per-step gathers
// are cheap ds_loads from the landed slot.
// ---------------------------------------------------------------------------
__device__ __forceinline__ void gather_lp(const float* rowlds, float lset,
                                          const int cls[9], const bool valid[9],
                                          float lp[9]) {
    #pragma unroll
    for (int i = 0; i < 9; ++i)
        lp[i] = valid[i] ? (rowlds[cls[i]] - lset) : NEGV;
}

__device__ __forceinline__ void ctc_update(float alpha[9], const float lp[9],
                                           const bool allow[9], int lane) {
    float p1[9], p2v[9];
    #pragma unroll
    for (int i = 0; i < 9; ++i) {
        float up1 = __shfl_up(alpha[i], 1, 32);
        float up2 = __shfl_up(alpha[i], 2, 32);
        float l31 = (i > 0) ? __shfl(alpha[i - 1], 31, 32) : NEGV;
        float l30 = (i > 0) ? __shfl(alpha[i - 1], 30, 32) : NEGV;
        float a1  = (lane == 0) ? l31 : up1;
        float a2  = (lane == 0) ? l30 : ((lane == 1) ? l31 : up2);
        p1[i]  = a1;
        p2v[i] = allow[i] ? a2 : NEGV;
    }
    #pragma unroll
    for (int i = 0; i < 9; ++i)
        alpha[i] = laddexp(laddexp(alpha[i], p1[i]), p2v[i]) + lp[i];
}

__global__ __launch_bounds__(32)
void ctc_scan_kernel(const float* __restrict__ preds,
                     const int*   __restrict__ labels,
                     const int*   __restrict__ in_len,
                     const int*   __restrict__ lab_len,
                     const float* __restrict__ lse,
                     float*       __restrict__ nll_ws) {
    const int n    = blockIdx.x;
    const int lane = threadIdx.x;

    __shared__ float ring[DEPTH][C_];          // 4 x 2KB emission-row ring
    __shared__ float sa[L_];

    // Per-lane extended-label metadata: s = i*32 + lane
    int  cls[9];
    bool valid[9];
    bool allow[9];
    #pragma unroll
    for (int i = 0; i < 9; ++i) {
        int s = i * 32 + lane;
        bool v = (s < L_);
        int  c = 0;
        bool ak = false;
        if (v && (s & 1)) {
            c = labels[n * S_ + ((s - 1) >> 1)];
            if (s >= 3) {
                int cp = labels[n * S_ + ((s - 3) >> 1)];
                ak = (c != cp);                 // skip allowed: non-blank, non-repeat
            }
        }
        cls[i] = c; valid[i] = v; allow[i] = ak;
    }

    auto issue_row = [&](int t) {
        AS1 const char* gp =
            (AS1 const char*)(preds + ((size_t)t * N_ + n) * C_) + lane * 16;
        AS3 char* lp = (AS3 char*)&ring[t & (DEPTH - 1)][0] + lane * 16;
        #pragma unroll
        for (int c = 0; c < 4; ++c)
            __builtin_amdgcn_global_load_async_to_lds_b128(
                (AS1 v4i*)(gp + c * 512), (AS3 v4i*)(lp + c * 512), 0, 0);
    };

    const int len = in_len[n];

    // t = 0 initialization (direct global gather): alpha0[s]=lp[s] for s<=1
    float alpha[9];
    {
        const float* row0 = preds + (size_t)n * C_;
        float ls0 = lse[n];
        #pragma unroll
        for (int i = 0; i < 9; ++i) {
            int s = i * 32 + lane;
            float lp0 = valid[i] ? (row0[cls[i]] - ls0) : NEGV;
            alpha[i] = (s <= 1) ? lp0 : NEGV;
        }
    }

    // Prime the ring with rows 1..DEPTH
    #pragma unroll
    for (int t = 1; t <= DEPTH; ++t) issue_row(t);

    // Main loop: wait-oldest, gather from LDS, recurse, refill slot.
    for (int t = 1; t <= T_ - DEPTH; ++t) {
        __builtin_amdgcn_s_wait_asynccnt(4 * (DEPTH - 1)); // row t landed
        const float* rowlds = &ring[t & (DEPTH - 1)][0];
        float lset = lse[t * N_ + n];
        float lp[9];
        gather_lp(rowlds, lset, cls, valid, lp);
        if (t < len) ctc_update(alpha, lp, allow, lane);    // uniform freeze
        asm volatile("s_wait_dscnt 0" ::: "memory");        // slot reads drained
        if (t + DEPTH < T_) issue_row(t + DEPTH);
    }
    // Tail: everything already issued; drain and finish.
    __builtin_amdgcn_s_wait_asynccnt(0);
    for (int t = T_ - DEPTH + 1; t < T_; ++t) {
        const float* rowlds = &ring[t & (DEPTH - 1)][0];
        float lset = lse[t * N_ + n];
        float lp[9];
        gather_lp(rowlds, lset, cls, valid, lp);
        if (t < len) ctc_update(alpha, lp, allow, lane);
    }

    // Tail: nll = -logaddexp(alpha[2*ll], alpha[2*ll-1])
    #pragma unroll
    for (int i = 0; i < 9; ++i) {
        int s = i * 32 + lane;
        if (s < L_) sa[s] = alpha[i];
    }
    __syncthreads();
    if (lane == 0) {
        int ll = lab_len[n];                    // in [64, 128]
        float a1 = sa[2 * ll];
        float a2 = sa[2 * ll - 1];
        nll_ws[n] = -laddexp(a1, a2);
    }
}

// ---------------------------------------------------------------------------
// Kernel 3: zero-infinity clamp, divide by label length, deterministic mean.
// ---------------------------------------------------------------------------
__global__ void ctc_final_kernel(const float* __restrict__ nll_ws,
                                 const int*   __restrict__ lab_len,
                                 float*       __restrict__ out) {
    if (threadIdx.x == 0 && blockIdx.x == 0) {
        float s = 0.0f;
        for (int n = 0; n < N_; ++n) {
            float v = nll_ws[n];
            v = (v < 1e6f) ? v : 0.0f;          // nll < -NEG*1e-3 keeps, else 0
            float dl = fmaxf((float)lab_len[n], 1.0f);
            s += v / dl;
        }
        out[0] = s / (float)N_;
    }
}

extern "C" void kernel_launch(void* const* d_in, const int* in_sizes, int n_in,
                              void* d_out, int out_size, void* d_ws, size_t ws_size,
                              hipStream_t stream) {
    const float* preds   = (const float*)d_in[0];   // (T, N, C) f32
    const int*   labels  = (const int*)d_in[1];     // (N, S) i32
    const int*   in_len  = (const int*)d_in[2];     // (N,) i32
    const int*   lab_len = (const int*)d_in[3];     // (N,) i32

    float* lse    = (float*)d_ws;                   // T*N floats = 256 KB
    float* nll_ws = lse + (size_t)T_ * N_;          // N floats

    hipLaunchKernelGGL(ctc_lse_kernel, dim3((T_ * N_) / 16), dim3(32), 0, stream,
                       preds, lse);
    hipLaunchKernelGGL(ctc_scan_kernel, dim3(N_), dim3(32), 0, stream,
                       preds, labels, in_len, lab_len, lse, nll_ws);
    hipLaunchKernelGGL(ctc_final_kernel, dim3(1), dim3(32), 0, stream,
                       nll_ws, lab_len, (float*)d_out);
}